// StructureEncoder_54030688584369
// MI455X (gfx1250) — compile-verified
//
#include <hip/hip_runtime.h>

typedef __attribute__((ext_vector_type(2))) float v2f;
typedef __attribute__((ext_vector_type(8))) float v8f;

#define NNODES 100000
#define NEDGES 1600000
#define FD 128
#define FH 128
#define FL 64
#define BN_EPS 1e-5f

// ---------------- degree ----------------
__global__ void deg_count_kernel(const long long* __restrict__ dst,
                                 float* __restrict__ deg) {
    int e = blockIdx.x * blockDim.x + threadIdx.x;
    if (e < NEDGES) atomicAdd(&deg[(int)dst[e]], 1.0f);
}

__global__ void deg_isqrt_kernel(float* __restrict__ deg) {
    int i = blockIdx.x * blockDim.x + threadIdx.x;
    if (i < NNODES) deg[i] = rsqrtf(deg[i] + 1.0f);  // +1 for self loop
}

// ---------------- GEMM: out[N x 128] = Hm[N x 128] @ W[128 x 128] ----------------
// One wave (32 lanes) computes a 16-row x 128-col strip using V_WMMA_F32_16X16X4_F32.
__global__ void gemm_nk128_kernel(const float* __restrict__ Hm,
                                  const float* __restrict__ W,
                                  float* __restrict__ out) {
    const int lane  = threadIdx.x & 31;
    const int wave  = threadIdx.x >> 5;
    const int strip = blockIdx.x * (blockDim.x >> 5) + wave;   // 16-row strip id
    const int row0  = strip * 16;
    if (row0 >= NNODES) return;           // wave-uniform: EXEC stays all-1s for WMMA

    const int m = lane & 15;              // row within tile / col within tile
    const int g = lane >> 4;              // lane-half group

    v8f acc[8];
#pragma unroll
    for (int t = 0; t < 8; ++t) acc[t] = (v8f)0.0f;

    const float* hrow = Hm + (size_t)(row0 + m) * FD;
    for (int kk = 0; kk < FD; kk += 4) {
        // A frag: 16x4 fp32; lane holds row m, K = kk + g*2 + {0,1}
        v2f a = *(const v2f*)(hrow + kk + g * 2);
#pragma unroll
        for (int t = 0; t < 8; ++t) {
            // B frag: 4x16 fp32; VGPR v holds K = kk + g*2 + v, N = m
            const int col = t * 16 + m;
            v2f b;
            b.x = W[(size_t)(kk + g * 2 + 0) * FH + col];
            b.y = W[(size_t)(kk + g * 2 + 1) * FH + col];
            acc[t] = __builtin_amdgcn_wmma_f32_16x16x4_f32(
                false, a, false, b, (short)0, acc[t], false, false);
        }
    }
    // C/D layout: VGPR r, lanes0-15 -> M=r; lanes16-31 -> M=r+8; N = m
#pragma unroll
    for (int t = 0; t < 8; ++t)
#pragma unroll
        for (int r = 0; r < 8; ++r)
            out[(size_t)(row0 + r + g * 8) * FH + t * 16 + m] = acc[t][r];
}

// ---------------- edge scatter: agg[dst] += hW[src] * isq[src]*isq[dst] ----------------
__global__ void edge_scatter_kernel(const float* __restrict__ hW,
                                    const long long* __restrict__ src,
                                    const long long* __restrict__ dstv,
                                    const float* __restrict__ isq,
                                    float* __restrict__ agg) {
    long long idx = (long long)blockIdx.x * blockDim.x + threadIdx.x;
    if (idx >= (long long)NEDGES * 32) return;
    const int e = (int)(idx >> 5);
    const int c = ((int)idx & 31) * 4;
    const int s = (int)src[e];
    const int d = (int)dstv[e];
    const float norm = isq[s] * isq[d];
    const float4 v = *(const float4*)(hW + (size_t)s * FH + c);
    float* ap = agg + (size_t)d * FH + c;
    atomicAdd(ap + 0, v.x * norm);
    atomicAdd(ap + 1, v.y * norm);
    atomicAdd(ap + 2, v.z * norm);
    atomicAdd(ap + 3, v.w * norm);
}

// ---------------- self-loop + bias + BN stats (fused single pass) ----------------
// thread t = channel; grid-stride over rows. stats[0:128]=sum, stats[128:256]=sumsq
__global__ void finalize_bnstats_kernel(float* __restrict__ agg,
                                        const float* __restrict__ hW,
                                        const float* __restrict__ isq,
                                        const float* __restrict__ bias,
                                        float* __restrict__ stats) {
    const int c = threadIdx.x;  // 128 threads
    const float b = bias[c];
    float s = 0.f, s2 = 0.f;
    for (int row = blockIdx.x; row < NNODES; row += gridDim.x) {
        const float di = isq[row];
        const size_t off = (size_t)row * FH + c;
        const float v = agg[off] + hW[off] * (di * di) + b;
        agg[off] = v;
        s += v;
        s2 += v * v;
    }
    atomicAdd(&stats[c], s);
    atomicAdd(&stats[FH + c], s2);
}

// ---------------- BN normalize + ReLU ----------------
__global__ void bn_relu_kernel(float* __restrict__ h,
                               const float* __restrict__ stats,
                               const float* __restrict__ gamma,
                               const float* __restrict__ beta) {
    long long i = (long long)blockIdx.x * blockDim.x + threadIdx.x;
    if (i >= (long long)NNODES * FH) return;
    const int c = (int)(i & (FH - 1));
    const float inv_n = 1.0f / (float)NNODES;
    const float mean = stats[c] * inv_n;
    const float var  = stats[FH + c] * inv_n - mean * mean;
    const float v = (h[i] - mean) * rsqrtf(var + BN_EPS) * gamma[c] + beta[c];
    h[i] = fmaxf(v, 0.0f);
}

// ---------------- heads: mu = h2@Wmu + bmu ; lv = h2@Wlv + blv (shared A frags) ----------------
__global__ void heads_kernel(const float* __restrict__ h2,
                             const float* __restrict__ Wmu, const float* __restrict__ bmu,
                             const float* __restrict__ Wlv, const float* __restrict__ blv,
                             float* __restrict__ mu, float* __restrict__ lv) {
    const int lane  = threadIdx.x & 31;
    const int wave  = threadIdx.x >> 5;
    const int strip = blockIdx.x * (blockDim.x >> 5) + wave;
    const int row0  = strip * 16;
    if (row0 >= NNODES) return;

    const int m = lane & 15;
    const int g = lane >> 4;

    v8f am[4], al[4];
#pragma unroll
    for (int t = 0; t < 4; ++t) { am[t] = (v8f)0.0f; al[t] = (v8f)0.0f; }

    const float* hrow = h2 + (size_t)(row0 + m) * FH;
    for (int kk = 0; kk < FH; kk += 4) {
        v2f a = *(const v2f*)(hrow + kk + g * 2);
#pragma unroll
        for (int t = 0; t < 4; ++t) {
            const int col = t * 16 + m;
            v2f bm, bl;
            bm.x = Wmu[(size_t)(kk + g * 2 + 0) * FL + col];
            bm.y = Wmu[(size_t)(kk + g * 2 + 1) * FL + col];
            bl.x = Wlv[(size_t)(kk + g * 2 + 0) * FL + col];
            bl.y = Wlv[(size_t)(kk + g * 2 + 1) * FL + col];
            am[t] = __builtin_amdgcn_wmma_f32_16x16x4_f32(
                false, a, false, bm, (short)0, am[t], false, false);
            al[t] = __builtin_amdgcn_wmma_f32_16x16x4_f32(
                false, a, false, bl, (short)0, al[t], false, false);
        }
    }
#pragma unroll
    for (int t = 0; t < 4; ++t) {
        const int col = t * 16 + m;
        const float bmv = bmu[col];
        const float blv_ = blv[col];
#pragma unroll
        for (int r = 0; r < 8; ++r) {
            const size_t row = (size_t)(row0 + r + g * 8);
            mu[row * FL + col] = am[t][r] + bmv;
            lv[row * FL + col] = al[t][r] + blv_;
        }
    }
}

// ---------------- launch ----------------
extern "C" void kernel_launch(void* const* d_in, const int* in_sizes, int n_in,
                              void* d_out, int out_size, void* d_ws, size_t ws_size,
                              hipStream_t stream) {
    (void)in_sizes; (void)n_in; (void)out_size; (void)ws_size;

    const float*      x   = (const float*)d_in[0];
    const long long*  ei  = (const long long*)d_in[1];   // int64 edge_index [2, E]
    const long long*  src = ei;
    const long long*  dst = ei + NEDGES;
    const float* W1  = (const float*)d_in[2];
    const float* b1  = (const float*)d_in[3];
    const float* g1  = (const float*)d_in[4];
    const float* be1 = (const float*)d_in[5];
    const float* W2  = (const float*)d_in[6];
    const float* b2  = (const float*)d_in[7];
    const float* g2  = (const float*)d_in[8];
    const float* be2 = (const float*)d_in[9];
    const float* Wmu = (const float*)d_in[10];
    const float* bmu = (const float*)d_in[11];
    const float* Wlv = (const float*)d_in[12];
    const float* blv = (const float*)d_in[13];

    float* ws    = (float*)d_ws;
    float* deg   = ws;                                  // N  (becomes deg_isqrt)
    float* stats = ws + NNODES;                         // 256
    float* bufA  = stats + 256;                         // N*128
    float* bufB  = bufA + (size_t)NNODES * FH;          // N*128
    float* h1    = (float*)d_out;                       // N*128 scratch; overwritten by mu/lv
    float* out_mu = (float*)d_out;                      // N*64
    float* out_lv = out_mu + (size_t)NNODES * FL;       // N*64

    const size_t big_bytes = (size_t)NNODES * FH * sizeof(float);

    // degrees (in-degree + self loop) -> rsqrt
    hipMemsetAsync(deg, 0, NNODES * sizeof(float), stream);
    deg_count_kernel<<<(NEDGES + 255) / 256, 256, 0, stream>>>(dst, deg);
    deg_isqrt_kernel<<<(NNODES + 255) / 256, 256, 0, stream>>>(deg);

    const int strips      = NNODES / 16;                   // 6250 (N % 16 == 0)
    const int gemm_blocks = (strips + 7) / 8;              // 8 waves/block
    const long long sthreads = (long long)NEDGES * 32;
    const int scat_blocks = (int)((sthreads + 255) / 256);
    const long long n_el  = (long long)NNODES * FH;
    const int elem_blocks = (int)((n_el + 255) / 256);

    // ---- layer 1 ----
    gemm_nk128_kernel<<<gemm_blocks, 256, 0, stream>>>(x, W1, bufA);
    hipMemsetAsync(h1, 0, big_bytes, stream);
    hipMemsetAsync(stats, 0, 256 * sizeof(float), stream);
    edge_scatter_kernel<<<scat_blocks, 256, 0, stream>>>(bufA, src, dst, deg, h1);
    finalize_bnstats_kernel<<<1024, 128, 0, stream>>>(h1, bufA, deg, b1, stats);
    bn_relu_kernel<<<elem_blocks, 256, 0, stream>>>(h1, stats, g1, be1);

    // ---- layer 2 ----
    gemm_nk128_kernel<<<gemm_blocks, 256, 0, stream>>>(h1, W2, bufB);
    hipMemsetAsync(bufA, 0, big_bytes, stream);
    hipMemsetAsync(stats, 0, 256 * sizeof(float), stream);
    edge_scatter_kernel<<<scat_blocks, 256, 0, stream>>>(bufB, src, dst, deg, bufA);
    finalize_bnstats_kernel<<<1024, 128, 0, stream>>>(bufA, bufB, deg, b2, stats);
    bn_relu_kernel<<<elem_blocks, 256, 0, stream>>>(bufA, stats, g2, be2);

    // ---- heads (read bufA = h2, write d_out) ----
    heads_kernel<<<gemm_blocks, 256, 0, stream>>>(bufA, Wmu, bmu, Wlv, blv, out_mu, out_lv);
}